// ConvolutionLayer_40699110097302
// MI455X (gfx1250) — compile-verified
//
#include <hip/hip_runtime.h>
#include <hip/hip_bf16.h>

// ---------------------------------------------------------------------------
// Spectral conv layer, frequency-domain restructure for MI455X (gfx1250).
//   out[b,p,l] = irfft_f( sum_k ev_k^.25 * k_f[k,f] * sum_d M[k,d,p]*u_f[b,d,f] )
// Pipeline: rfft(u) -> rfft(filters)/N -> per-k WMMA GEMM + complex scale -> irfft.
// GEMM uses V_WMMA_F32_16X16X4_F32 (fp32 in / fp32 acc) on wave32.
// u_f stored as float4 quads {Re(2d2), Re(2d2+1), Im(2d2), Im(2d2+1)} so one
// global_load_b128 feeds the Re+Im WMMA pair of a d-step.
// ---------------------------------------------------------------------------

#define B_SZ   4
#define D_IN   256
#define D_OUT  256
#define L_SEQ  2048
#define K_NUM  24
#define FFT_N  4096
#define FFT_LG 12
#define NBIN   2049          // rfft bins of a 4096-pt transform
#define FP     2304          // padded freq dim = 9*256 (multiple of 256)
#define FBLK   256           // f-window per block (8 waves x 2 tiles x 16)
#define FFT_T  256           // threads per FFT block (8 waves)

typedef float v2f __attribute__((ext_vector_type(2)));
typedef float v8f __attribute__((ext_vector_type(8)));

__device__ __forceinline__ float2 cmulf(float2 a, float2 w) {
  return make_float2(a.x * w.x - a.y * w.y, a.x * w.y + a.y * w.x);
}

// In-LDS Stockham radix-2 FFT, natural-order output after FFT_LG (even) stages,
// so the result lands back in the buffer passed as `x`. Unnormalized.
template <int SIGN>
__device__ __forceinline__ void stockham_fft(float2* x, float2* y, int tid) {
#pragma unroll 1
  for (int stage = 0; stage < FFT_LG; ++stage) {
    const int n = FFT_N >> stage;
    const int m = n >> 1;
    const int s = 1 << stage;
    const float theta = (float)SIGN * 6.28318530717958647692f / (float)n;
#pragma unroll 1
    for (int t = tid; t < FFT_N / 2; t += FFT_T) {
      const int p = t >> stage;        // 0..m-1
      const int q = t & (s - 1);       // 0..s-1
      const float2 a = x[q + s * p];
      const float2 b = x[q + s * (p + m)];
      const float2 sum = make_float2(a.x + b.x, a.y + b.y);
      const float2 dif = make_float2(a.x - b.x, a.y - b.y);
      float wi, wr;
      __sincosf(theta * (float)p, &wi, &wr);
      y[q + s * (2 * p)]     = sum;
      y[q + s * (2 * p + 1)] = cmulf(dif, make_float2(wr, wi));
    }
    __syncthreads();
    float2* tmp = x; x = y; y = tmp;
  }
}

// Forward rfft of real rows (length L_SEQ, zero-padded to FFT_N).
// QUAD=true : u rows, row = b*256+d; writes quad layout
//             out4[((b*128 + d/2)*FP + f)*4 + {d&1, 2+(d&1)}] = {Re, Im}
// QUAD=false: filter rows; writes plain planes outRe/outIm of stride FP.
// Pad tail [NBIN, FP) is zero-filled in both modes.
template <bool QUAD>
__global__ __launch_bounds__(FFT_T)
void fft_fwd_rows(const float* __restrict__ in, float scale,
                  float* __restrict__ out4,
                  float* __restrict__ outRe, float* __restrict__ outIm) {
  __shared__ float2 bufA[FFT_N];
  __shared__ float2 bufB[FFT_N];
  const int row = blockIdx.x;
  const int tid = threadIdx.x;
  const float* src = in + (size_t)row * L_SEQ;
  for (int i = tid; i < FFT_N; i += FFT_T)
    bufA[i] = make_float2((i < L_SEQ) ? src[i] : 0.0f, 0.0f);
  __syncthreads();
  stockham_fft<-1>(bufA, bufB, tid);

  if (QUAD) {
    const int b = row >> 8;          // D_IN == 256
    const int d = row & 255;
    float* o = out4 + (((size_t)b * (D_IN / 2) + (d >> 1)) * FP) * 4 + (d & 1);
    for (int i = tid; i < FP; i += FFT_T) {
      float re = 0.0f, im = 0.0f;
      if (i < NBIN) { re = scale * bufA[i].x; im = scale * bufA[i].y; }
      o[(size_t)i * 4]     = re;     // Re component slot
      o[(size_t)i * 4 + 2] = im;     // Im component slot
    }
  } else {
    float* oR = outRe + (size_t)row * FP;
    float* oI = outIm + (size_t)row * FP;
    for (int i = tid; i < FP; i += FFT_T) {
      float re = 0.0f, im = 0.0f;
      if (i < NBIN) { re = scale * bufA[i].x; im = scale * bufA[i].y; }
      oR[i] = re;
      oI[i] = im;
    }
  }
}

// Inverse: rebuild the conj-symmetric 4096-pt spectrum from NBIN bins, run the
// unnormalized inverse FFT (matches irfft(norm='forward')), emit Re of first L.
__global__ __launch_bounds__(FFT_T)
void ifft_rows_real(const float* __restrict__ Gr, const float* __restrict__ Gi,
                    float* __restrict__ out) {
  __shared__ float2 bufA[FFT_N];
  __shared__ float2 bufB[FFT_N];
  const int row = blockIdx.x;
  const int tid = threadIdx.x;
  const float* gr = Gr + (size_t)row * FP;
  const float* gi = Gi + (size_t)row * FP;
  for (int i = tid; i < FFT_N; i += FFT_T) {
    const int  f   = (i <= FFT_N / 2) ? i : (FFT_N - i);
    const float sg = (i <= FFT_N / 2) ? 1.0f : -1.0f;
    bufA[i] = make_float2(gr[f], sg * gi[f]);
  }
  __syncthreads();
  stockham_fft<1>(bufA, bufB, tid);
  float* o = out + (size_t)row * L_SEQ;
  for (int i = tid; i < L_SEQ; i += FFT_T) o[i] = bufA[i].x;
}

// ---------------------------------------------------------------------------
// Frequency-domain contraction with V_WMMA_F32_16X16X4_F32.
// Block = 256 threads (8 wave32): blockIdx = (f-window of 256, p-tile of 16, b).
// Wave w owns two f-tiles at fb0 + 32w and fb0 + 32w + 16, sharing one
// A-fragment per d-step. Per k: stage M[k,:,p0:p0+16] into LDS as A[d][p],
// run 64 d-steps of {1 ds_load_2addr + 2 global_load_b128 + 4 WMMA}, then
// apply the per-lane complex scale ev_k^.25 * k_f[k,f] in the epilogue
// (scale is lane-constant because the C layout puts N=f in the lane index).
// ---------------------------------------------------------------------------
__global__ __launch_bounds__(256)
void spectral_contract_wmma(const float4* __restrict__ U4,
                            const float* __restrict__ kfr, const float* __restrict__ kfi,
                            const float* __restrict__ Mw, const float* __restrict__ eigvals,
                            float* __restrict__ Gr, float* __restrict__ Gi) {
  __shared__ float Ms[D_IN * 16];   // 16 KB: A-tile, [d][p_local]
  __shared__ float kfs[2][FBLK];    // current filter spectrum over block's f-window

  const int tid  = threadIdx.x;
  const int wave = tid >> 5;
  const int lane = tid & 31;
  const int m    = lane & 15;       // row-in-half / N index
  const int g    = lane >> 4;       // lane half
  const int fb0  = blockIdx.x * FBLK;
  const int p0   = blockIdx.y * 16;
  const int b    = blockIdx.z;
  const int f1l  = wave * 32 + m;   // local f of tile 1
  const int f2l  = f1l + 16;        // local f of tile 2
  const int f1   = fb0 + f1l;
  const int f2   = fb0 + f2l;

  const float4* u4 = U4 + (size_t)b * (D_IN / 2) * FP;

  v8f gR1 = {}, gI1 = {}, gR2 = {}, gI2 = {};

  for (int k = 0; k < K_NUM; ++k) {
    __syncthreads();
    // Stage A = M[k]^T tile: Ms[d][p_local] = M[k, d, p0 + p_local].
    for (int i = tid; i < D_IN * 4; i += 256) {      // 1024 float4 moves
      const int d  = i >> 2;
      const int p4 = (i & 3) * 4;
      *(float4*)&Ms[d * 16 + p4] =
          *(const float4*)(Mw + ((size_t)k * D_IN + d) * D_OUT + p0 + p4);
    }
    kfs[0][tid] = kfr[(size_t)k * FP + fb0 + tid];
    kfs[1][tid] = kfi[(size_t)k * FP + fb0 + tid];
    __syncthreads();

    const float evq = sqrtf(sqrtf(eigvals[k]));      // ev^0.25, no __powf blob
    const float sr1 = evq * kfs[0][f1l], si1 = evq * kfs[1][f1l];
    const float sr2 = evq * kfs[0][f2l], si2 = evq * kfs[1][f2l];

    v8f tR1 = {}, tI1 = {}, tR2 = {}, tI2 = {};
#pragma unroll 4
    for (int d0 = 0; d0 < D_IN; d0 += 4) {
      const int dA = d0 + 2 * g;                     // A cols 2g, 2g+1
      const int d2 = (d0 >> 1) + g;                  // quad row = dA/2
      v2f a;
      a.x = Ms[dA * 16 + m];
      a.y = Ms[(dA + 1) * 16 + m];
      const float4 q1 = u4[(size_t)d2 * FP + f1];    // {r0, r1, i0, i1}
      const float4 q2 = u4[(size_t)d2 * FP + f2];
      v2f bR1 = {q1.x, q1.y}, bI1 = {q1.z, q1.w};
      v2f bR2 = {q2.x, q2.y}, bI2 = {q2.z, q2.w};
      // (neg_a, A, neg_b, B, c_mod, C, reuse_a, reuse_b)
      tR1 = __builtin_amdgcn_wmma_f32_16x16x4_f32(false, a, false, bR1, (short)0, tR1, false, false);
      tI1 = __builtin_amdgcn_wmma_f32_16x16x4_f32(false, a, false, bI1, (short)0, tI1, false, false);
      tR2 = __builtin_amdgcn_wmma_f32_16x16x4_f32(false, a, false, bR2, (short)0, tR2, false, false);
      tI2 = __builtin_amdgcn_wmma_f32_16x16x4_f32(false, a, false, bI2, (short)0, tI2, false, false);
    }
    // G += (sr + i*si) * (tR + i*tI) for both f-tiles.
#pragma unroll
    for (int j = 0; j < 8; ++j) {
      gR1[j] += sr1 * tR1[j] - si1 * tI1[j];
      gI1[j] += sr1 * tI1[j] + si1 * tR1[j];
      gR2[j] += sr2 * tR2[j] - si2 * tI2[j];
      gI2[j] += sr2 * tI2[j] + si2 * tR2[j];
    }
  }

  // C/D layout: VGPR j holds M = j (lanes 0-15) / j+8 (lanes 16-31), N = m.
  float* grp = Gr + (size_t)b * D_OUT * FP;
  float* gip = Gi + (size_t)b * D_OUT * FP;
#pragma unroll
  for (int j = 0; j < 8; ++j) {
    const int p = p0 + 8 * g + j;
    grp[(size_t)p * FP + f1] = gR1[j];
    gip[(size_t)p * FP + f1] = gI1[j];
    grp[(size_t)p * FP + f2] = gR2[j];
    gip[(size_t)p * FP + f2] = gI2[j];
  }
}

// ---------------------------------------------------------------------------
// Workspace layout (floats):
//   U4 : [B][D_IN/2][FP][4]  = 4*128*2304*4 = 18.9 MB
//   kfr, kfi : [K][FP]
//   Gr, Gi   : [B*D_OUT][FP] = 18.9 MB
// Total ~38.2 MB.
// ---------------------------------------------------------------------------
extern "C" void kernel_launch(void* const* d_in, const int* in_sizes, int n_in,
                              void* d_out, int out_size, void* d_ws, size_t ws_size,
                              hipStream_t stream) {
  const float* u    = (const float*)d_in[0];   // [B, D_IN, L]
  const float* ev   = (const float*)d_in[1];   // [K]
  const float* evec = (const float*)d_in[2];   // [K, L]
  const float* Mw   = (const float*)d_in[3];   // [K, D_IN, D_OUT]
  float* out = (float*)d_out;                  // [B, D_OUT, L]

  float* ws = (float*)d_ws;
  const size_t rows = (size_t)B_SZ * D_IN;     // 1024 (== B*D_OUT)
  float* U4f = ws;                                         // quad u_f
  float* kfr = U4f + (size_t)B_SZ * (D_IN / 2) * FP * 4;
  float* kfi = kfr + (size_t)K_NUM * FP;
  float* Gr  = kfi + (size_t)K_NUM * FP;
  float* Gi  = Gr  + rows * FP;

  fft_fwd_rows<true><<<dim3((unsigned)rows), dim3(FFT_T), 0, stream>>>(
      u, 1.0f, U4f, nullptr, nullptr);
  fft_fwd_rows<false><<<dim3(K_NUM), dim3(FFT_T), 0, stream>>>(
      evec, 1.0f / (float)FFT_N, nullptr, kfr, kfi);
  spectral_contract_wmma<<<dim3(FP / FBLK, D_OUT / 16, B_SZ), dim3(256), 0, stream>>>(
      (const float4*)U4f, kfr, kfi, Mw, ev, Gr, Gi);
  ifft_rows_real<<<dim3((unsigned)rows), dim3(FFT_T), 0, stream>>>(Gr, Gi, out);
}